// PostProcess_46823733461734
// MI455X (gfx1250) — compile-verified
//
#include <hip/hip_runtime.h>
#include <hip/hip_bf16.h>

#define BATCH   256
#define QN      900
#define CN      91
#define NSEL    300
#define NFLAT   (QN * CN)          // 81900 (divisible by 4)
#define THREADS 256
#define SORTN   512

typedef __attribute__((ext_vector_type(2))) float v2f;
typedef __attribute__((ext_vector_type(8))) float v8f;

// Pointee types for the async global->LDS builtin (param type revealed by
// clang diagnostic: 'int __vector(4) __device__ *' = AS1 ptr to int32x4).
typedef int vi4 __attribute__((vector_size(16)));
typedef __attribute__((address_space(1))) vi4 gvi4;   // global int32x4
typedef __attribute__((address_space(3))) vi4 lvi4;   // LDS int32x4

// ---- gfx1250 async global->LDS path (probe-guarded) -------------------------
#if __has_builtin(__builtin_amdgcn_global_load_async_to_lds_b128)
#define HAVE_ASYNC 1
#define ASYNC_LOAD_B128(gp, lp) \
    __builtin_amdgcn_global_load_async_to_lds_b128((gvi4*)(gp), (lvi4*)(lp), 0, 0)
#else
#define HAVE_ASYNC 0
#endif

#if __has_builtin(__builtin_amdgcn_s_wait_asynccnt)
#define WAIT_ASYNC(n) __builtin_amdgcn_s_wait_asynccnt(n)
#else
#define WAIT_ASYNC(n) asm volatile("s_wait_asynccnt %0" :: "i"(n) : "memory")
#endif

// Monotonic order-preserving key: larger float <-> larger unsigned key
__device__ __forceinline__ unsigned key_of(float x) {
    unsigned u = __float_as_uint(x);
    return (u & 0x80000000u) ? ~u : (u | 0x80000000u);
}
__device__ __forceinline__ float val_of(unsigned k) {
    unsigned u = (k & 0x80000000u) ? (k & 0x7fffffffu) : ~k;
    return __uint_as_float(u);
}

// Two-level descending threshold search over `nb` buckets:
// finds bucket T s.t. count(buckets > T) < need <= count(buckets >= T).
__device__ __forceinline__ void find_threshold(unsigned* hist, unsigned* gsum,
                                               int nb, unsigned need,
                                               volatile unsigned* s_bucket,
                                               volatile unsigned* s_krem, int tid)
{
    const int g = nb / THREADS;               // 8 or 4 buckets per thread
    unsigned s = 0;
    #pragma unroll
    for (int k = 0; k < 8; ++k)               // max group size is 8
        if (k < g) s += hist[tid * g + k];
    gsum[tid] = s;
    __syncthreads();
    if (tid == 0) {
        unsigned acc = 0; int gsel = 0;
        for (int gi = THREADS - 1; gi >= 0; --gi) {
            unsigned c = gsum[gi];
            if (acc + c >= need) { gsel = gi; break; }
            acc += c;
        }
        unsigned rem = need - acc;
        for (int bkt = gsel * g + g - 1; bkt >= gsel * g; --bkt) {
            unsigned c = hist[bkt];
            if (c >= rem) { *s_bucket = (unsigned)bkt; *s_krem = rem; break; }
            rem -= c;
        }
    }
    __syncthreads();
}

__global__ __launch_bounds__(THREADS)
void postprocess_topk_kernel(const float* __restrict__ logits,
                             const float* __restrict__ boxes_in,
                             const float* __restrict__ tsizes,
                             float* __restrict__ out)
{
    __shared__ unsigned hist[2048];
    __shared__ unsigned gsum[THREADS];
    __shared__ unsigned sel_key[SORTN];
    __shared__ unsigned sel_idx[SORTN];
    __shared__ __attribute__((aligned(16))) float stage[2][THREADS * 4]; // 8 KB
    __shared__ unsigned s_bucket, s_krem, s_cnt_gt, s_cnt_eq;

    const int b   = blockIdx.x;
    const int tid = threadIdx.x;
    const float*  row  = logits + (size_t)b * NFLAT;
    const float4* row4 = reinterpret_cast<const float4*>(row);

    // ============ Phase 1: exact 32-bit radix select of the threshold key ====
    unsigned prefix = 0;
    unsigned krem   = NSEL;

    // ---- Pass 0: bits [31:21] (2048 buckets). HBM-bound scan: stage through
    // LDS with double-buffered async loads (ASYNCcnt) when available. --------
    for (int i = tid; i < 2048; i += THREADS) hist[i] = 0u;
    __syncthreads();

#if HAVE_ASYNC
    {
        constexpr int CHUNK = THREADS * 4;              // 1024 floats / chunk
        const int nfull = NFLAT / CHUNK;                // 79 full chunks
        ASYNC_LOAD_B128(row + tid * 4, &stage[0][tid * 4]);
        for (int c = 0; c < nfull; ++c) {
            const int buf = c & 1;
            if (c + 1 < nfull) {
                ASYNC_LOAD_B128(row + (size_t)(c + 1) * CHUNK + tid * 4,
                                &stage[buf ^ 1][tid * 4]);
                WAIT_ASYNC(1);
            } else {
                WAIT_ASYNC(0);
            }
            float4 v = *(const float4*)&stage[buf][tid * 4];
            atomicAdd(&hist[(key_of(v.x) >> 21) & 2047u], 1u);
            atomicAdd(&hist[(key_of(v.y) >> 21) & 2047u], 1u);
            atomicAdd(&hist[(key_of(v.z) >> 21) & 2047u], 1u);
            atomicAdd(&hist[(key_of(v.w) >> 21) & 2047u], 1u);
        }
        for (int i4 = (nfull * CHUNK) / 4 + tid; i4 < NFLAT / 4; i4 += THREADS) {
            float4 v = row4[i4];
            atomicAdd(&hist[(key_of(v.x) >> 21) & 2047u], 1u);
            atomicAdd(&hist[(key_of(v.y) >> 21) & 2047u], 1u);
            atomicAdd(&hist[(key_of(v.z) >> 21) & 2047u], 1u);
            atomicAdd(&hist[(key_of(v.w) >> 21) & 2047u], 1u);
        }
    }
#else
    for (int i4 = tid; i4 < NFLAT / 4; i4 += THREADS) {
        __builtin_prefetch(row4 + i4 + THREADS, 0, 0);  // global_prefetch_b8
        float4 v = row4[i4];
        atomicAdd(&hist[(key_of(v.x) >> 21) & 2047u], 1u);
        atomicAdd(&hist[(key_of(v.y) >> 21) & 2047u], 1u);
        atomicAdd(&hist[(key_of(v.z) >> 21) & 2047u], 1u);
        atomicAdd(&hist[(key_of(v.w) >> 21) & 2047u], 1u);
    }
#endif
    __syncthreads();
    find_threshold(hist, gsum, 2048, krem, &s_bucket, &s_krem, tid);
    prefix = s_bucket;
    krem   = s_krem;
    __syncthreads();

    // ---- Passes 1,2: bits [20:10] then [9:0]; L2-resident re-scans ----------
    {
        const int pass_bits[2]  = {11, 10};
        const int pass_shift[2] = {10, 0};
        const int pass_done[2]  = {11, 22};
        for (int p = 0; p < 2; ++p) {
            const int bits  = pass_bits[p];
            const int shift = pass_shift[p];
            const int dbits = pass_done[p];
            const int nb    = 1 << bits;
            const unsigned mask = (unsigned)(nb - 1);
            for (int i = tid; i < nb; i += THREADS) hist[i] = 0u;
            __syncthreads();
            for (int i4 = tid; i4 < NFLAT / 4; i4 += THREADS) {
                float4 v = row4[i4];
                unsigned ks[4] = {key_of(v.x), key_of(v.y), key_of(v.z), key_of(v.w)};
                #pragma unroll
                for (int k2 = 0; k2 < 4; ++k2) {
                    unsigned k = ks[k2];
                    if ((k >> (32 - dbits)) == prefix)
                        atomicAdd(&hist[(k >> shift) & mask], 1u);
                }
            }
            __syncthreads();
            find_threshold(hist, gsum, nb, krem, &s_bucket, &s_krem, tid);
            prefix = (prefix << bits) | s_bucket;
            krem   = s_krem;
            __syncthreads();
        }
    }

    const unsigned K    = prefix;        // exact threshold key
    const unsigned n_gt = NSEL - krem;   // count strictly greater than K (exact)

    // ============ Phase 2: compact winners into LDS ===========================
    if (tid == 0) { s_cnt_gt = 0u; s_cnt_eq = 0u; }
    for (int i = tid; i < SORTN; i += THREADS) { sel_key[i] = 0u; sel_idx[i] = 0u; }
    __syncthreads();

    for (int i4 = tid; i4 < NFLAT / 4; i4 += THREADS) {
        float4 v = row4[i4];
        unsigned ks[4] = {key_of(v.x), key_of(v.y), key_of(v.z), key_of(v.w)};
        #pragma unroll
        for (int k2 = 0; k2 < 4; ++k2) {
            unsigned k = ks[k2];
            if (k > K) {
                unsigned pos = atomicAdd(&s_cnt_gt, 1u);     // < n_gt by construction
                sel_key[pos] = k; sel_idx[pos] = (unsigned)(4 * i4 + k2);
            } else if (k == K) {
                unsigned e = atomicAdd(&s_cnt_eq, 1u);
                if (e < krem) {
                    unsigned pos = n_gt + e;
                    sel_key[pos] = k; sel_idx[pos] = (unsigned)(4 * i4 + k2);
                }
            }
        }
    }
    __syncthreads();

    // ============ Phase 3: bitonic sort 512 slots, descending by key =========
    for (unsigned size = 2; size <= SORTN; size <<= 1) {
        for (unsigned stride = size >> 1; stride >= 1; stride >>= 1) {
            __syncthreads();
            unsigned t = (unsigned)tid;                      // 256 pairs
            unsigned a = 2u * t - (t & (stride - 1u));
            unsigned c = a + stride;
            bool desc = ((a & size) == 0u);
            unsigned ka = sel_key[a], kc = sel_key[c];
            if ((ka < kc) == desc) {
                sel_key[a] = kc; sel_key[c] = ka;
                unsigned ia = sel_idx[a]; sel_idx[a] = sel_idx[c]; sel_idx[c] = ia;
            }
        }
    }
    __syncthreads();

    // ============ Phase 4: scores and labels ==================================
    float* out_scores = out;
    float* out_labels = out + (size_t)BATCH * NSEL;
    float* out_boxes  = out + (size_t)2 * BATCH * NSEL;

    for (int j = tid; j < NSEL; j += THREADS) {
        float lg = val_of(sel_key[j]);
        out_scores[(size_t)b * NSEL + j] = 1.0f / (1.0f + __expf(-lg));
        out_labels[(size_t)b * NSEL + j] = (float)(sel_idx[j] % CN);
    }
    __syncthreads();

    // ============ Phase 5: box transform via V_WMMA_F32_16X16X4_F32 ===========
    // xyxy_scaled(16x16, cols 0..3 valid) = A(16x4: cx,cy|w,h) x B(4x16),
    // scale [w,h,w,h] folded into B:  B[:,n] = M[:,n] * scale[n]
    //   M = [[1,0,1,0],[0,1,0,1],[-.5,0,.5,0],[0,-.5,0,.5]]
    const int lane = tid & 31;
    const int wave = tid >> 5;            // 8 waves
    const float img_h = tsizes[2 * b + 0];
    const float img_w = tsizes[2 * b + 1];

    v2f bmat; bmat.x = 0.0f; bmat.y = 0.0f;
    {
        int n = lane & 15;
        if (n < 4) {
            float s = (n & 1) ? img_h : img_w;
            if (lane < 16) {                     // rows K=0 (cx), K=1 (cy)
                bmat.x = (n == 0 || n == 2) ? s : 0.0f;
                bmat.y = (n == 1 || n == 3) ? s : 0.0f;
            } else {                             // rows K=2 (w), K=3 (h)
                bmat.x = (n == 0) ? -0.5f * s : ((n == 2) ? 0.5f * s : 0.0f);
                bmat.y = (n == 1) ? -0.5f * s : ((n == 3) ? 0.5f * s : 0.0f);
            }
        }
    }

    const int NT = (NSEL + 15) / 16;      // 19 tiles of 16 boxes
    for (int tile = wave; tile < NT; tile += 8) {
        const int m  = lane & 15;
        const int j  = tile * 16 + m;
        const int jj = (j < NSEL) ? j : (NSEL - 1);       // clamped -> always valid
        const unsigned idx = sel_idx[jj];
        const int q = (int)(idx / CN);
        // Unconditional aligned 8B load, then select-zero padding lanes.
        const v2f bv = *(const v2f*)(boxes_in + (((size_t)b * QN + q) * 4)
                                     + ((lane < 16) ? 0 : 2));
        const bool valid = (j < NSEL);
        v2f amat;
        amat.x = valid ? bv.x : 0.0f;     // lanes 0-15: cx,cy ; 16-31: w,h
        amat.y = valid ? bv.y : 0.0f;

#if __has_builtin(__builtin_amdgcn_wmma_f32_16x16x4_f32)
        v8f acc = {};
        acc = __builtin_amdgcn_wmma_f32_16x16x4_f32(
                  false, amat, false, bmat, (short)0, acc, false, false);
        // D layout: lane L<16 -> D[v][L]; lane L>=16 -> D[v+8][L-16]
        const int n = lane & 15;
        if (n < 4) {
            const int row0 = (lane < 16) ? 0 : 8;
            float* dst = out_boxes + (size_t)(b * NSEL + tile * 16 + row0) * 4 + n;
            if (tile * 16 + 16 <= NSEL) {           // full tile: no bounds checks
                #pragma unroll
                for (int v = 0; v < 8; ++v) dst[v * 4] = acc[v];
            } else {                                 // single partial tile
                #pragma unroll
                for (int v = 0; v < 8; ++v)
                    if (tile * 16 + row0 + v < NSEL) dst[v * 4] = acc[v];
            }
        }
#else
        if (valid && lane < 16) {
            const float* bq = boxes_in + (((size_t)b * QN + q) * 4);
            float cx = bq[0], cy = bq[1], w = bq[2], h = bq[3];
            size_t base = (size_t)(b * NSEL + j) * 4;
            out_boxes[base + 0] = (cx - 0.5f * w) * img_w;
            out_boxes[base + 1] = (cy - 0.5f * h) * img_h;
            out_boxes[base + 2] = (cx + 0.5f * w) * img_w;
            out_boxes[base + 3] = (cy + 0.5f * h) * img_h;
        }
#endif
    }
}

extern "C" void kernel_launch(void* const* d_in, const int* in_sizes, int n_in,
                              void* d_out, int out_size, void* d_ws, size_t ws_size,
                              hipStream_t stream) {
    const float* logits = (const float*)d_in[0];   // (B, Q, C) f32
    const float* boxes  = (const float*)d_in[1];   // (B, Q, 4) f32
    const float* tsz    = (const float*)d_in[2];   // (B, 2)    f32 (h, w)
    float* out = (float*)d_out;                    // scores | labels | boxes

    postprocess_topk_kernel<<<BATCH, THREADS, 0, stream>>>(logits, boxes, tsz, out);
}